// TrainNhppNm_6098853560631
// MI455X (gfx1250) — compile-verified
//
#include <hip/hip_runtime.h>
#include <math.h>

typedef __attribute__((ext_vector_type(2))) float v2f;
typedef __attribute__((ext_vector_type(8))) float v8f;

#define N_NODES    512
#define TILE       16
#define NTILES     32          // 512 / 16
#define NPAIRTILES 528         // 32*33/2  (ti <= tj)
#define NS         10          // quadrature samples per interval
#define NT_Q       40          // B*S quadrature time points
#define NT         120         // + B*E event time points
#define NTASKS     (NT * NPAIRTILES)   // 63360 wave-tasks

// One wave = one 16x16 pair tile at one time point.
// Gram tile via V_WMMA_F32_16X16X4_F32 (K = D = 4 exactly).
__global__ __launch_bounds__(256)
void nhpp_tile_kernel(const float* __restrict__ gamma,
                      const float* __restrict__ z0,          // [3][512][4]
                      const float* __restrict__ t_init,      // [4]
                      const float* __restrict__ t_last,      // [4]
                      const float* __restrict__ event_times, // [4*20]
                      float* __restrict__ block_partials)
{
    __shared__ float warp_sums[8];

    const int lane = threadIdx.x & 31;
    const int warp = threadIdx.x >> 5;
    const int half = lane >> 4;      // 0: lanes 0-15, 1: lanes 16-31
    const int m    = lane & 15;
    const int wavesInGrid = (gridDim.x * blockDim.x) >> 5;

    const float4* z0v = (const float4*)z0;   // [3][512] rows of D=4

    float acc_signed = 0.0f;   // += w*exp(ll) (quadrature),  -= ll (events)

    for (int task = (blockIdx.x * blockDim.x + threadIdx.x) >> 5;
         task < NTASKS; task += wavesInGrid) {

        const int time_idx = task / NPAIRTILES;
        const int pair_idx = task - time_idx * NPAIRTILES;

        // Closed-form decode of (ti <= tj): reverse index into a 1..32 triangle.
        // q = 527 - p ; u = floor((sqrt(8q+1)-1)/2) ; ti = 31-u ; tj = 31-(q-u(u+1)/2)
        const int q = (NPAIRTILES - 1) - pair_idx;
        int u = (int)((__builtin_amdgcn_sqrtf((float)(8 * q + 1)) - 1.0f) * 0.5f);
        if (u * (u + 1) / 2 > q) --u;                    // float-floor fixups
        if ((u + 1) * (u + 2) / 2 <= q) ++u;
        const int ti = (NTILES - 1) - u;
        const int tj = (NTILES - 1) - (q - u * (u + 1) / 2);

        // time point + weight (wave-uniform)
        float t, w = 0.0f;
        bool isEvent;
        if (time_idx < NT_Q) {
            const int b = time_idx / NS;
            const int s = time_idx - b * NS;
            const float step = (t_last[b] - t_init[b]) * (1.0f / NS);
            t = t_init[b] + step * (float)s;
            w = step;
            isEvent = false;
        } else {
            t = event_times[time_idx - NT_Q];
            isEvent = true;
        }

        const int irow = ti * TILE + m;
        const int jrow = tj * TILE + m;
        const float c2 = 0.5f * t * t;   // ORDER=3: coeffs 1, t, t^2/2

        const float4 zi0 = z0v[irow], zi1 = z0v[N_NODES + irow], zi2 = z0v[2*N_NODES + irow];
        const float4 zj0 = z0v[jrow], zj1 = z0v[N_NODES + jrow], zj2 = z0v[2*N_NODES + jrow];

        const float zix = zi0.x + t*zi1.x + c2*zi2.x;
        const float ziy = zi0.y + t*zi1.y + c2*zi2.y;
        const float ziz = zi0.z + t*zi1.z + c2*zi2.z;
        const float ziw = zi0.w + t*zi1.w + c2*zi2.w;
        const float zjx = zj0.x + t*zj1.x + c2*zj2.x;
        const float zjy = zj0.y + t*zj1.y + c2*zj2.y;
        const float zjz = zj0.z + t*zj1.z + c2*zj2.z;
        const float zjw = zj0.w + t*zj1.w + c2*zj2.w;

        // A frag (16x4, M = lane%16, K = 2*half + v) and
        // B frag (4x16, N = lane%16, K = v + 2*half): identical per-lane gather.
        v2f afrag, bfrag;
        afrag[0] = half ? ziz : zix;
        afrag[1] = half ? ziw : ziy;
        bfrag[0] = half ? zjz : zjx;
        bfrag[1] = half ? zjw : zjy;

        const float sqi_own = zix*zix + ziy*ziy + ziz*ziz + ziw*ziw;
        const float gi_own  = gamma[irow];
        const float sqj     = zjx*zjx + zjy*zjy + zjz*zjz + zjw*zjw;
        const float gj      = gamma[jrow];

        v8f c = {0.f,0.f,0.f,0.f,0.f,0.f,0.f,0.f};
        c = __builtin_amdgcn_wmma_f32_16x16x4_f32(
                /*neg_a=*/false, afrag, /*neg_b=*/false, bfrag,
                /*c_mod=*/(short)0, c, /*reuse_a=*/false, /*reuse_b=*/false);

        // Broadcast the 8 needed i-row scalars (lane M owns row M's values).
        float sqi_r[8], gi_r[8];
#pragma unroll
        for (int r = 0; r < 8; ++r) {
            const int M = half * 8 + r;
            sqi_r[r] = __shfl(sqi_own, M, 32);
            gi_r[r]  = __shfl(gi_own,  M, 32);
        }

        // Strict upper triangle: include element r iff r < lim (per-lane const).
        const int lim = jrow - ti * TILE - half * 8;

        float ps0 = 0.0f, ps1 = 0.0f;
        if (isEvent) {
#pragma unroll
            for (int r = 0; r < 8; ++r) {
                const float d2   = sqi_r[r] + sqj - 2.0f * c[r];
                const float dist = __builtin_amdgcn_sqrtf(fmaxf(d2, 0.0f));
                const float ll   = gi_r[r] + gj - dist;
                const float msk  = (r < lim) ? 1.0f : 0.0f;
                if (r & 1) ps1 = fmaf(msk, -ll, ps1);    // un-sinkable mask
                else       ps0 = fmaf(msk, -ll, ps0);
            }
        } else {
#pragma unroll
            for (int r = 0; r < 8; ++r) {
                const float d2   = sqi_r[r] + sqj - 2.0f * c[r];
                const float dist = __builtin_amdgcn_sqrtf(fmaxf(d2, 0.0f));
                const float el   = w * __expf(gi_r[r] + gj - dist);
                const float msk  = (r < lim) ? 1.0f : 0.0f;
                if (r & 1) ps1 = fmaf(msk, el, ps1);
                else       ps0 = fmaf(msk, el, ps0);
            }
        }
        acc_signed += ps0 + ps1;
    }

    // wave32 reduce
#pragma unroll
    for (int off = 16; off > 0; off >>= 1)
        acc_signed += __shfl_xor(acc_signed, off, 32);
    if (lane == 0) warp_sums[warp] = acc_signed;
    __syncthreads();
    if (threadIdx.x == 0) {
        float s = 0.0f;
#pragma unroll
        for (int i = 0; i < 8; ++i) s += warp_sums[i];
        block_partials[blockIdx.x] = s;
    }
}

// Deterministic fixed-order final reduction: NLL = sum(partials)
__global__ __launch_bounds__(256)
void nhpp_reduce_kernel(const float* __restrict__ partials, int n,
                        float* __restrict__ out)
{
    __shared__ float lds[256];
    float s = 0.0f;
    for (int i = threadIdx.x; i < n; i += 256) s += partials[i];
    lds[threadIdx.x] = s;
    __syncthreads();
    if (threadIdx.x == 0) {
        float tot = 0.0f;
        for (int i = 0; i < 256; ++i) tot += lds[i];
        out[0] = tot;
    }
}

extern "C" void kernel_launch(void* const* d_in, const int* in_sizes, int n_in,
                              void* d_out, int out_size, void* d_ws, size_t ws_size,
                              hipStream_t stream) {
    (void)in_sizes; (void)n_in; (void)out_size;
    const float* gamma       = (const float*)d_in[0];
    const float* z0          = (const float*)d_in[1];
    const float* t_init      = (const float*)d_in[2];
    const float* t_last      = (const float*)d_in[3];
    const float* event_times = (const float*)d_in[4];
    // d_in[5]/d_in[6] (pair_i/pair_j) are canonical triu(k=1) indices; the
    // kernel enumerates the same i<j set implicitly via tile coordinates.
    float* out = (float*)d_out;
    float* ws  = (float*)d_ws;

    int nblk = 2048;                            // 16384 waves, ~3.9 tasks/wave
    size_t cap = ws_size / sizeof(float);
    if ((size_t)nblk > cap) nblk = (int)cap;
    if (nblk < 1) nblk = 1;

    nhpp_tile_kernel<<<nblk, 256, 0, stream>>>(gamma, z0, t_init, t_last,
                                               event_times, ws);
    nhpp_reduce_kernel<<<1, 256, 0, stream>>>(ws, nblk, out);
}